// RelationshipModule_57612691308661
// MI455X (gfx1250) — compile-verified
//
#include <hip/hip_runtime.h>
#include <hip/hip_bf16.h>

// CDNA5 / gfx1250 WMMA kernel for the bilinear "relationship module":
//   out[b,x,y,X,Y,o] = sum_{i,j} inp[b,x,y,i] * W[dx,dy,o,i,j] * inp[b,X,Y,j] + bias[dx,dy,o]
// reformulated as a single GEMM per 16-row chunk:  out[n,o] = sum_{kk=(i,j)} A[n,kk] * Wmat[kk,o]
// with A[n,kk] = inp1[n,i]*inp2[n,j] built on the fly (one v_pk_mul_f32 per v_wmma_f32_16x16x4_f32).
// W is staged to LDS with GLOBAL_LOAD_ASYNC_TO_LDS_B64 (ASYNCcnt path, VGPR-bypass gather-scatter).

typedef float v2f __attribute__((ext_vector_type(2)));
typedef float v4f __attribute__((ext_vector_type(4)));
typedef float v8f __attribute__((ext_vector_type(8)));

#define NB  128   // batch
#define NS  8     // spatial
#define NF  32    // features
#define NO  16    // output channels == WMMA N
#define ND  15    // 2*S-1 relative offsets

__device__ __forceinline__ int cdim(int d) {
    int t = d - 7;
    if (t < 0) t = -t;
    return 8 - t;                 // number of valid x for offset d
}

__global__ __launch_bounds__(256)
void relmod_wmma_f32_kernel(const float* __restrict__ inp,
                            const float* __restrict__ W,
                            const float* __restrict__ bias,
                            float* __restrict__ out)
{
    // Wmat[kk,o] staged as pairs: dword[(p*16 + o)*2 + c] = W[dx,dy,o, (2p+c)>>5, (2p+c)&31]
    // -> B-fragment for k-step s is one ds_load_b64 at lane*8 + s*256 (16-bit imm), conflict-free.
    __shared__ float ldsW[16384];  // 64 KB

    // ---- uniform map: blockIdx.x -> (dx, dy, local block) ; grid = sum c(dx)*c(dy) = 4096 ----
    int rem = (int)blockIdx.x;
    int dx = 0, dy = 0;
    for (int t = 0; t < ND * ND; ++t) {
        int tx = t / ND, ty = t % ND;
        int n = cdim(tx) * cdim(ty);
        if (rem < n) { dx = tx; dy = ty; break; }
        rem -= n;
    }
    const int cx = cdim(dx), cy = cdim(dy);
    const int lb = rem;                       // local block in [0, cx*cy)

    const int tid  = (int)threadIdx.x;
    const int lane = tid & 31;
    const int wid  = tid >> 5;                // 8 waves per block
    const int h    = lane >> 4;               // half-wave (K-pair selector)
    const int o    = lane & 15;               // output channel / A-row id

    // ---- async VGPR-bypass LDS fill of Wmat (gather-scatter, tracked by ASYNCcnt) ----
    // Thread handles pairs P2 = tid + 256k, k = 0..31:
    //   global byte offset advances +128 per k (o2,j0 fixed, i += 1)
    //   LDS byte dest     advances +2048 per k (P2*8)
    const float* Wblk = W + (size_t)(dx * ND + dy) * (NO * NF * NF);
    {
        int o2 = tid & 15;
        int p0 = tid >> 4;
        unsigned goff  = (unsigned)((o2 * (NF * NF) + (p0 >> 4) * NF + ((2 * p0) & 31)) * 4);
        unsigned laddr = (unsigned)(size_t)(&ldsW[tid * 2]);
        #pragma unroll 1
        for (int k = 0; k < 32; ++k) {
            asm volatile("global_load_async_to_lds_b64 %0, %1, %2"
                         :: "v"(laddr), "v"(goff), "s"(Wblk) : "memory");
            goff  += 128u;
            laddr += 2048u;
        }
    }

    // ---- this wave's 16-row chunk: rows n0..n0+15 over valid (b, x, y) ----
    const int chunk = lb * 8 + wid;           // in [0, 8*cx*cy), exact — no idle waves
    const int n0 = chunk * 16;
    const int x0 = (7 - dx) > 0 ? (7 - dx) : 0;
    const int y0 = (7 - dy) > 0 ? (7 - dy) : 0;

    // decode this lane's A-row m = o
    unsigned nm = (unsigned)(n0 + o);
    unsigned yl = nm % (unsigned)cy; unsigned t1 = nm / (unsigned)cy;
    unsigned xl = t1 % (unsigned)cx; unsigned bb = t1 / (unsigned)cx;
    int x = (int)xl + x0, y = (int)yl + y0;
    int X = x + dx - 7,  Y = y + dy - 7;

    const float* r1 = inp + (size_t)(((int)bb * NS + x) * NS + y) * NF;          // inp1 row (i)
    const float* r2 = inp + (size_t)(((int)bb * NS + X) * NS + Y) * NF + 2 * h;  // inp2 row (j), half-phase

    // inp1: all 32 features in VGPRs (constant-indexed under full unroll) — overlaps async DMA
    float a1e[32];
    #pragma unroll
    for (int t = 0; t < 8; ++t) {
        v4f v = *(const v4f*)(r1 + 4 * t);
        a1e[4 * t + 0] = v.x; a1e[4 * t + 1] = v.y;
        a1e[4 * t + 2] = v.z; a1e[4 * t + 3] = v.w;
    }
    // inp2: the 16 j-values this half-wave feeds: j = 4k + 2h + {0,1}
    v2f q2[8];
    #pragma unroll
    for (int k = 0; k < 8; ++k)
        q2[k] = *(const v2f*)(r2 + 4 * k);

    // all of this wave's async LDS writes landed -> barrier makes every wave's W tile visible
    asm volatile("s_wait_asynccnt 0x0" ::: "memory");
    __syncthreads();

    // ---- main GEMM: 16x16 out tile, K = 1024 = 256 steps of v_wmma_f32_16x16x4_f32 ----
    const v2f* bp = (const v2f*)ldsW;         // pair table; lane base + 16-bit imm offsets
    v8f acc = {0.f, 0.f, 0.f, 0.f, 0.f, 0.f, 0.f, 0.f};
    #pragma unroll
    for (int s = 0; s < 256; ++s) {
        float av = a1e[s >> 3];               // i = kk>>5 = s>>3 (constant index)
        v2f  q  = q2[s & 7];                  // (inp2[j0], inp2[j0+1]) (constant index)
        v2f  af; af.x = av * q.x; af.y = av * q.y;        // A frag: outer-product on the fly
        v2f  bf = bp[lane + 32 * s];          // ds_load_b64, imm offset s*256 <= 65280
        acc = __builtin_amdgcn_wmma_f32_16x16x4_f32(
                  false, af, false, bf, (short)0, acc, false, false);
    }

    // ---- epilogue: bias + coalesced stores (C layout: vgpr k, lane l -> M=8h+k, N=o) ----
    float bv = bias[(dx * ND + dy) * NO + o];
    #pragma unroll
    for (int k = 0; k < 8; ++k) {
        unsigned nk = (unsigned)(n0 + 8 * h + k);
        unsigned yl2 = nk % (unsigned)cy; unsigned t2 = nk / (unsigned)cy;
        unsigned xl2 = t2 % (unsigned)cx; unsigned b2 = t2 / (unsigned)cx;
        int x2 = (int)xl2 + x0, y2 = (int)yl2 + y0;
        int X2 = x2 + dx - 7,  Y2 = y2 + dy - 7;
        size_t oi = (((((size_t)b2 * NS + x2) * NS + y2) * NS + X2) * NS + Y2) * NO + o;
        out[oi] = acc[k] + bv;
    }
}

extern "C" void kernel_launch(void* const* d_in, const int* in_sizes, int n_in,
                              void* d_out, int out_size, void* d_ws, size_t ws_size,
                              hipStream_t stream) {
    const float* inp  = (const float*)d_in[0];  // (128,8,8,32)
    const float* W    = (const float*)d_in[1];  // (15,15,16,32,32)
    const float* bias = (const float*)d_in[2];  // (15,15,16)
    float* out = (float*)d_out;                 // (128,8,8,8,8,16)
    (void)in_sizes; (void)n_in; (void)out_size; (void)d_ws; (void)ws_size;

    // grid = sum_{dx,dy} c(dx)*c(dy) = 64*64/... = 4096 blocks, 8 waves each,
    // one 16-row x 16-o output tile per wave; covers every output element exactly once.
    relmod_wmma_f32_kernel<<<dim3(4096), dim3(256), 0, stream>>>(inp, W, bias, out);
}